// FineGenerator_40793599378107
// MI455X (gfx1250) — compile-verified
//
#include <hip/hip_runtime.h>
#include <cmath>

// ---------------------------------------------------------------------------
// Contextual attention for MI455X (gfx1250), fp32 WMMA path.
//   B=8, C=128, H=W=96, RATE=2 -> H2=W2=48, L=2304, K1=C*9=1152,
//   raw patches 4x4 -> M2=C*16=2048, output 96x96 per channel.
// ---------------------------------------------------------------------------

typedef float v2f __attribute__((ext_vector_type(2)));
typedef float v8f __attribute__((ext_vector_type(8)));
typedef int   v4i __attribute__((ext_vector_type(4)));
typedef __attribute__((address_space(1))) v4i* as1_v4i;
typedef __attribute__((address_space(3))) v4i* as3_v4i;

#define BATCH  8
#define CCH    128
#define HFULL  96
#define HS     48
#define LPATCH 2304      // 48*48
#define KDIM1  1152      // C*9
#define MDIM2  2048      // C*16
#define NPIX   9216      // 96*96

// Async global->LDS path (gfx1250 GLOBAL_LOAD_ASYNC_TO_LDS_B128, ASYNCcnt).
// Guarded: if the builtin is absent we fall back to register staging.
#if __has_builtin(__builtin_amdgcn_global_load_async_to_lds_b128)
#define ASYNC_LDS 1
#else
#define ASYNC_LDS 0
#endif

__device__ __forceinline__ void wait_asynccnt0() {
#if __has_builtin(__builtin_amdgcn_s_wait_asynccnt)
    __builtin_amdgcn_s_wait_asynccnt(0);
#else
    asm volatile("s_wait_asynccnt 0x0" ::: "memory");
#endif
}

#if ASYNC_LDS
__device__ __forceinline__ void async_copy16(const float* g, float* l) {
    __builtin_amdgcn_global_load_async_to_lds_b128(
        (as1_v4i)(v4i*)const_cast<float*>(g),
        (as3_v4i)(v4i*)l,
        0, 0);
}
#endif

// ------------------------- mask -> mm[2304] --------------------------------
__global__ void build_mm_kernel(const float* __restrict__ mask, float* __restrict__ mm) {
    int l = blockIdx.x * blockDim.x + threadIdx.x;
    if (l >= LPATCH) return;
    int i = l / HS, j = l % HS;
    float s = 0.f;
    for (int dh = -1; dh <= 1; ++dh)
        for (int dw = -1; dw <= 1; ++dw) {
            int a = i + dh, b = j + dw;
            if (a >= 0 && a < HS && b >= 0 && b < HS)
                s += mask[(size_t)(a * 8) * 384 + (size_t)b * 8];
        }
    mm[l] = (s == 0.f) ? 1.f : 0.f;
}

// ------------- Wn: L x K1, L2-normalized 3x3xC patches of b_s --------------
__global__ __launch_bounds__(128)
void build_wn_kernel(const float* __restrict__ b, float* __restrict__ Wn, int bidx) {
    const int l = blockIdx.x;            // patch index (i*48+j)
    const int i = l / HS, j = l % HS;
    const int t = threadIdx.x;           // 0..127, 9 elements each
    float v[9];
    float ss = 0.f;
#pragma unroll
    for (int r9 = 0; r9 < 9; ++r9) {
        int k  = t + r9 * 128;           // k = c*9 + kh*3 + kw
        int c  = k / 9, r = k % 9;
        int kh = r / 3, kw = r % 3;
        int ii = i - 1 + kh, jj = j - 1 + kw;
        float val = 0.f;
        if (ii >= 0 && ii < HS && jj >= 0 && jj < HS)
            val = b[((size_t)(bidx * CCH + c) * HFULL + 2 * ii) * HFULL + 2 * jj];
        v[r9] = val;
        ss += val * val;
    }
    __shared__ float red[128];
    red[t] = ss;
    __syncthreads();
    for (int s = 64; s > 0; s >>= 1) {
        if (t < s) red[t] += red[t + s];
        __syncthreads();
    }
    const float inv = 1.f / fmaxf(sqrtf(red[0]), 1e-4f);
#pragma unroll
    for (int r9 = 0; r9 < 9; ++r9) {
        int k = t + r9 * 128;
        Wn[(size_t)l * KDIM1 + k] = v[r9] * inv;
    }
}

// ------------- Fp: K1 x L, 3x3xC patches of f_s ----------------------------
__global__ void build_fp_kernel(const float* __restrict__ f, float* __restrict__ Fp, int bidx) {
    int idx = blockIdx.x * blockDim.x + threadIdx.x;   // < K1*L
    int k = idx / LPATCH, p = idx % LPATCH;
    int h = p / HS, w = p % HS;
    int c = k / 9, r = k % 9;
    int kh = r / 3, kw = r % 3;
    int hh = h - 1 + kh, ww = w - 1 + kw;
    float val = 0.f;
    if (hh >= 0 && hh < HS && ww >= 0 && ww < HS)
        val = f[((size_t)(bidx * CCH + c) * HFULL + 2 * hh) * HFULL + 2 * ww];
    Fp[idx] = val;
}

// ------------- RawT: M2 x L, flipped raw 4x4 patches of b ------------------
__global__ void build_rawt_kernel(const float* __restrict__ b, float* __restrict__ RawT, int bidx) {
    int idx = blockIdx.x * blockDim.x + threadIdx.x;   // < M2*L
    int m = idx / LPATCH, l = idx % LPATCH;
    int c = m >> 4, tap = m & 15;
    int kh = tap >> 2, kw = tap & 3;
    int i = l / HS, j = l % HS;
    int row = 2 * i + 2 - kh;
    int col = 2 * j + 2 - kw;
    float val = 0.f;
    if (row >= 0 && row < HFULL && col >= 0 && col < HFULL)
        val = b[((size_t)(bidx * CCH + c) * HFULL + row) * HFULL + col];
    RawT[idx] = val;
}

// ------------- generic fp32 WMMA GEMM: C = A(MxK) * B(KxN) -----------------
// 128x128 tile per 256-thread block (8 waves, each 32x64 = 2x4 fragments).
// Double-buffered LDS, one barrier per K-chunk. A panel staged via async
// global->LDS DMA (ASYNCcnt) when available; B panel register-staged and
// stored transposed so fragments are contiguous k-pairs. Row stride 20
// floats: 16B-aligned segments, 20*lane mod 64 sweeps 16 banks (no conflict).
// Requires M,N % 128 == 0, K % 16 == 0.
__global__ __launch_bounds__(256)
void gemm_wmma_f32_kernel(const float* __restrict__ A, const float* __restrict__ B,
                          float* __restrict__ C, int M, int N, int K) {
    __shared__ float As[2][128][20];
    __shared__ float BsT[2][128][20];
    const int t    = threadIdx.x;
    const int n0   = blockIdx.x * 128;
    const int m0   = blockIdx.y * 128;
    const int wave = t >> 5;             // 0..7
    const int lane = t & 31;
    const int wm   = (wave & 3) * 32;    // wave's M offset in tile
    const int wn   = (wave >> 2) * 64;   // wave's N offset in tile
    const int lm   = lane & 15;          // M (A) / N (B) index in fragment
    const int lk   = (lane >> 4) * 2;    // K base in fragment half

    v8f acc[2][4] = {};
    float4 pb[2];

    auto fetchB = [&](int kc) {
#pragma unroll
        for (int it = 0; it < 2; ++it) {
            int lin = t + it * 256;
            int br = lin >> 5, bn = lin & 31;      // B: 16 k-rows x 32 float4
            pb[it] = *reinterpret_cast<const float4*>(
                &B[(size_t)(kc + br) * N + n0 + bn * 4]);
        }
    };
    auto stageB = [&](int buf) {
#pragma unroll
        for (int it = 0; it < 2; ++it) {
            int lin = t + it * 256;
            int br = lin >> 5, bn = (lin & 31) * 4;
            BsT[buf][bn + 0][br] = pb[it].x;       // transposed: [n][k]
            BsT[buf][bn + 1][br] = pb[it].y;
            BsT[buf][bn + 2][br] = pb[it].z;
            BsT[buf][bn + 3][br] = pb[it].w;
        }
    };
#if ASYNC_LDS
    auto issueA = [&](int kc, int buf) {           // DMA 16B/lane, no VGPR staging
#pragma unroll
        for (int it = 0; it < 2; ++it) {
            int lin = t + it * 256;
            int ar = lin >> 2, as = (lin & 3) * 4; // A: 128 rows x 4 16B segs
            async_copy16(&A[(size_t)(m0 + ar) * K + kc + as], &As[buf][ar][as]);
        }
    };
#else
    float4 pa[2];
    auto fetchA = [&](int kc) {
#pragma unroll
        for (int it = 0; it < 2; ++it) {
            int lin = t + it * 256;
            int ar = lin >> 2, as = lin & 3;
            pa[it] = *reinterpret_cast<const float4*>(
                &A[(size_t)(m0 + ar) * K + kc + as * 4]);
        }
    };
    auto stageA = [&](int buf) {
#pragma unroll
        for (int it = 0; it < 2; ++it) {
            int lin = t + it * 256;
            int ar = lin >> 2, as = (lin & 3) * 4;
            *reinterpret_cast<float4*>(&As[buf][ar][as]) = pa[it]; // ds_store_b128
        }
    };
#endif
    auto compute = [&](int buf) {
#pragma unroll
        for (int k4 = 0; k4 < 4; ++k4) {
            const int kb = k4 * 4 + lk;
            v2f a[2], bf[4];
#pragma unroll
            for (int i = 0; i < 2; ++i) {
                a[i].x = As[buf][wm + i * 16 + lm][kb];
                a[i].y = As[buf][wm + i * 16 + lm][kb + 1];
            }
#pragma unroll
            for (int j = 0; j < 4; ++j) {
                bf[j].x = BsT[buf][wn + j * 16 + lm][kb];
                bf[j].y = BsT[buf][wn + j * 16 + lm][kb + 1];
            }
#pragma unroll
            for (int i = 0; i < 2; ++i)
#pragma unroll
                for (int j = 0; j < 4; ++j)
                    acc[i][j] = __builtin_amdgcn_wmma_f32_16x16x4_f32(
                        false, a[i], false, bf[j], (short)0, acc[i][j], false, false);
        }
    };

    const int nk = K / 16;
#if ASYNC_LDS
    fetchB(0);
    issueA(0, 0);
    stageB(0);
    wait_asynccnt0();
    __syncthreads();
    for (int i = 0; i < nk; ++i) {
        if (i + 1 < nk) {
            fetchB((i + 1) * 16);
            issueA((i + 1) * 16, (i + 1) & 1);   // DMA overlaps the WMMA chunk
        }
        compute(i & 1);
        if (i + 1 < nk) {
            stageB((i + 1) & 1);
            wait_asynccnt0();
            __syncthreads();
        }
    }
#else
    fetchB(0);
    fetchA(0);
    stageA(0);
    stageB(0);
    __syncthreads();
    for (int i = 0; i < nk; ++i) {
        if (i + 1 < nk) { fetchA((i + 1) * 16); fetchB((i + 1) * 16); }
        compute(i & 1);
        if (i + 1 < nk) {
            stageA((i + 1) & 1);
            stageB((i + 1) & 1);
            __syncthreads();
        }
    }
#endif

    // C/D layout: vgpr v, lanes 0-15 -> M=v, lanes 16-31 -> M=v+8; N=lane%16.
    const int mrow = m0 + wm + ((lane >> 4) << 3);
    const int ncol = n0 + wn + lm;
#pragma unroll
    for (int i = 0; i < 2; ++i)
#pragma unroll
        for (int j = 0; j < 4; ++j)
#pragma unroll
            for (int v = 0; v < 8; ++v)
                C[(size_t)(mrow + i * 16 + v) * N + ncol + j * 16] = acc[i][j][v];
}

// ------------- fused fuse-conv: both diagonal 3-sum passes in one ----------
// pass1: Z[r,c] = sum_d1 Y[r+d1, c+d1] (flat bounds); pass2 under the
// involution t(i) = (i%48)*48 + i/48:  out[r,c] = sum_d2 Z[t(t(r)+d2), t(t(c)+d2)].
__device__ __forceinline__ int tperm(int i) { return (i % HS) * HS + i / HS; }

__global__ void fuse_kernel(const float* __restrict__ in, float* __restrict__ out) {
    size_t idx = (size_t)blockIdx.x * blockDim.x + threadIdx.x;  // < L*L
    int r = (int)(idx / LPATCH), c = (int)(idx % LPATCH);
    int tr = tperm(r), tc = tperm(c);
    float s = 0.f;
#pragma unroll
    for (int d2 = -1; d2 <= 1; ++d2) {
        int jr = tr + d2, jc = tc + d2;
        if (jr < 0 || jr >= LPATCH || jc < 0 || jc >= LPATCH) continue;
        int rr = tperm(jr), cc = tperm(jc);
#pragma unroll
        for (int d1 = -1; d1 <= 1; ++d1) {
            int r1 = rr + d1, c1 = cc + d1;
            if (r1 >= 0 && r1 < LPATCH && c1 >= 0 && c1 < LPATCH)
                s += in[(size_t)r1 * LPATCH + c1];
        }
    }
    out[idx] = s;
}

// ------------- masked column softmax over l (axis 0), online ---------------
__global__ __launch_bounds__(256)
void softmax_col_kernel(const float* __restrict__ Y, const float* __restrict__ mm,
                        float* __restrict__ P) {
    const int p = blockIdx.x * 256 + threadIdx.x;  // < L
    float mx = -1e30f, sum = 0.f;
    for (int l = 0; l < LPATCH; ++l) {
        float s = 10.f * mm[l] * Y[(size_t)l * LPATCH + p];
        float nm = fmaxf(mx, s);
        sum = sum * expf(mx - nm) + expf(s - nm);
        mx = nm;
    }
    const float inv = 1.f / sum;
    for (int l = 0; l < LPATCH; ++l) {
        float s = 10.f * mm[l] * Y[(size_t)l * LPATCH + p];
        P[(size_t)l * LPATCH + p] = mm[l] * expf(s - mx) * inv;
    }
}

// ------------- final parity gather (transposed-conv epilogue) --------------
__global__ void out_gather_kernel(const float* __restrict__ U, float* __restrict__ out, int bidx) {
    int idx = blockIdx.x * blockDim.x + threadIdx.x;  // < C*NPIX
    int c = idx / NPIX;
    int pix = idx % NPIX;
    int y = pix / HFULL, x = pix % HFULL;
    float s = 0.f;
#pragma unroll
    for (int kh = 0; kh < 4; ++kh) {
        int ny = y + kh - 2;
        if (ny < 0 || (ny & 1)) continue;
        int h = ny >> 1;
        if (h >= HS) continue;
#pragma unroll
        for (int kw = 0; kw < 4; ++kw) {
            int nx = x + kw - 2;
            if (nx < 0 || (nx & 1)) continue;
            int w = nx >> 1;
            if (w >= HS) continue;
            s += U[(size_t)(c * 16 + kh * 4 + kw) * LPATCH + h * HS + w];
        }
    }
    out[(size_t)(bidx * CCH + c) * NPIX + pix] = 0.25f * s;
}

// ---------------------------------------------------------------------------
extern "C" void kernel_launch(void* const* d_in, const int* in_sizes, int n_in,
                              void* d_out, int out_size, void* d_ws, size_t ws_size,
                              hipStream_t stream) {
    (void)in_sizes; (void)n_in; (void)out_size; (void)ws_size;
    const float* f    = (const float*)d_in[0];   // (8,128,96,96)
    const float* b    = (const float*)d_in[1];   // (8,128,96,96)
    const float* mask = (const float*)d_in[2];   // (8,1,384,384)
    float* out = (float*)d_out;                  // (8,128,96,96)
    float* ws  = (float*)d_ws;

    // workspace layout (floats); ~79 MiB total, reused across samples
    float* mm   = ws;                                   // 2304 (rounded 4096)
    float* Wn   = ws + 4096;                            // L  x K1
    float* Fp   = Wn + (size_t)LPATCH * KDIM1;          // K1 x L
    float* RawT = Wn;                                   // M2 x L (reuses Wn+Fp)
    float* Ybuf = Fp + (size_t)KDIM1 * LPATCH;          // L x L
    float* Zbuf = Ybuf + (size_t)LPATCH * LPATCH;       // L x L
    float* Ubuf = Zbuf + (size_t)LPATCH * LPATCH;       // M2 x L

    build_mm_kernel<<<LPATCH / 256, 256, 0, stream>>>(mask, mm);

    for (int s = 0; s < BATCH; ++s) {
        build_wn_kernel<<<LPATCH, 128, 0, stream>>>(b, Wn, s);
        build_fp_kernel<<<(KDIM1 * LPATCH) / 256, 256, 0, stream>>>(f, Fp, s);
        // scores: Y = Wn (L x K1) * Fp (K1 x L)
        gemm_wmma_f32_kernel<<<dim3(LPATCH / 128, LPATCH / 128), 256, 0, stream>>>(
            Wn, Fp, Ybuf, LPATCH, LPATCH, KDIM1);
        fuse_kernel<<<((size_t)LPATCH * LPATCH) / 256, 256, 0, stream>>>(Ybuf, Zbuf);
        softmax_col_kernel<<<LPATCH / 256, 256, 0, stream>>>(Zbuf, mm, Ybuf);
        build_rawt_kernel<<<(MDIM2 * LPATCH) / 256, 256, 0, stream>>>(b, RawT, s);
        // deconv as GEMM: U = RawT (M2 x L) * P (L x L)
        gemm_wmma_f32_kernel<<<dim3(LPATCH / 128, MDIM2 / 128), 256, 0, stream>>>(
            RawT, Ybuf, Ubuf, MDIM2, LPATCH, LPATCH);
        out_gather_kernel<<<(CCH * NPIX) / 256, 256, 0, stream>>>(Ubuf, out, s);
    }
}